// PointNetFeatureInter_49237505082104
// MI455X (gfx1250) — compile-verified
//
#include <hip/hip_runtime.h>

// Problem constants (fixed by the reference setup_inputs()).
#define B_  8
#define N_  8192
#define S_  2048
#define D_  256
#define CHUNK 2048          // xyz1 points staged per LDS chunk (3*2048*4 = 24 KB)

typedef __attribute__((ext_vector_type(2))) float v2f;
typedef __attribute__((ext_vector_type(8))) float v8f;

// Branchless sorted-top3 insert: 3 compares + 6 selects, no EXEC divergence.
__device__ __forceinline__ void top3_insert(float d, int i,
                                            float& d0, int& i0,
                                            float& d1, int& i1,
                                            float& d2, int& i2) {
  const bool c2 = d < d2;
  const bool c1 = d < d1;
  const bool c0 = d < d0;
  const float nd2 = c1 ? d1 : (c2 ? d : d2);
  const int   ni2 = c1 ? i1 : (c2 ? i : i2);
  const float nd1 = c0 ? d0 : (c1 ? d : d1);
  const int   ni1 = c0 ? i0 : (c1 ? i : i1);
  const float nd0 = c0 ? d : d0;
  const int   ni0 = c0 ? i : i0;
  d2 = nd2; i2 = ni2;
  d1 = nd1; i1 = ni1;
  d0 = nd0; i0 = ni0;
}

// ---------------------------------------------------------------------------
// Kernel 1: squared distances via V_WMMA_F32_16X16X4_F32 + streaming top-3.
//
// D[m][n] = sum_k A[m][k]*B[k][n] + C[m][n]
//   A[m][:] = ( x1x, x1y, x1z, |x1_m|^2 )      (M = 16 points of current tile)
//   B[:][n] = (-2*x2x, -2*x2y, -2*x2z, 1 )     (N = the wave's 16 queries)
//   C[m][n] = |x2_n|^2
// => D[m][n] = squared distance(point m, query n).
//
// C/D layout (wave32): lane L holds column n = L%16; VGPR j holds row
// m = (L>=16 ? 8 : 0) + j.  Each lane keeps a private top-3 for ONE query;
// the two half-waves cover disjoint point rows and merge once at the end
// with __shfl_xor(...,16).
// ---------------------------------------------------------------------------
__global__ __launch_bounds__(256)
void knn3_wmma_kernel(const float* __restrict__ xyz1,
                      const float* __restrict__ xyz2,
                      float* __restrict__ wbuf,
                      int* __restrict__ ibuf) {
  __shared__ float lx[CHUNK], ly[CHUNK], lz[CHUNK];

  const int tid  = threadIdx.x;
  const int wave = tid >> 5;
  const int lane = tid & 31;
  const int n16  = lane & 15;
  const bool hi  = (lane >= 16);

  // This wave owns 16 consecutive queries; the whole block (128 queries)
  // stays inside one batch b since S_ % 128 == 0.
  const int gw    = blockIdx.x * 8 + wave;
  const int qflat = gw * 16;
  const int b     = qflat / S_;
  const int s0    = qflat % S_;
  const int s     = s0 + n16;

  // --- B operand + C operand: per-query, loop invariant -------------------
  const float qx = xyz2[(size_t)b * 3 * S_ + 0 * (size_t)S_ + s];
  const float qy = xyz2[(size_t)b * 3 * S_ + 1 * (size_t)S_ + s];
  const float qz = xyz2[(size_t)b * 3 * S_ + 2 * (size_t)S_ + s];
  const float q2 = qx * qx + qy * qy + qz * qz;

  v2f bq;
  bq.x = hi ? (-2.0f * qz) : (-2.0f * qx);
  bq.y = hi ? 1.0f         : (-2.0f * qy);

  v8f cmat;
#pragma unroll
  for (int j = 0; j < 8; ++j) cmat[j] = q2;

  float d0 = 3.4e38f, d1 = 3.4e38f, d2 = 3.4e38f;
  int   i0 = 0,       i1 = 0,       i2 = 0;

  const float* x1b = xyz1 + (size_t)b * 3 * N_;

  for (int c = 0; c < N_ / CHUNK; ++c) {
    // Stage one chunk of xyz1 into LDS (one global read per block).
    for (int off = tid; off < CHUNK; off += 256) {
      lx[off] = x1b[0 * (size_t)N_ + c * CHUNK + off];
      ly[off] = x1b[1 * (size_t)N_ + c * CHUNK + off];
      lz[off] = x1b[2 * (size_t)N_ + c * CHUNK + off];
    }
    __syncthreads();

#pragma unroll 2
    for (int t = 0; t < CHUNK / 16; ++t) {
      const int pl = t * 16 + n16;
      const float ax = lx[pl];
      const float ay = ly[pl];
      const float az = lz[pl];

      v2f am;
      am.x = hi ? az : ax;
      am.y = hi ? (ax * ax + ay * ay + az * az) : ay;

      // (neg_a, A, neg_b, B, c_mod, C, reuse_a, reuse_b)
      v8f dd = __builtin_amdgcn_wmma_f32_16x16x4_f32(
          false, am, false, bq, (short)0, cmat, false, false);

      const int gbase = c * CHUNK + t * 16 + (hi ? 8 : 0);
#pragma unroll
      for (int j = 0; j < 8; ++j)
        top3_insert(dd[j], gbase + j, d0, i0, d1, i1, d2, i2);
    }
    __syncthreads();
  }

  // Merge the two half-wave partial top-3 lists (other half's list arrives
  // sorted ascending; sequential insert preserves correctness).
  {
    float od; int oi;
    od = __shfl_xor(d0, 16); oi = __shfl_xor(i0, 16);
    top3_insert(od, oi, d0, i0, d1, i1, d2, i2);
    od = __shfl_xor(d1, 16); oi = __shfl_xor(i1, 16);
    top3_insert(od, oi, d0, i0, d1, i1, d2, i2);
    od = __shfl_xor(d2, 16); oi = __shfl_xor(i2, 16);
    top3_insert(od, oi, d0, i0, d1, i1, d2, i2);
  }

  if (!hi) {
    const float r0 = 1.0f / (d0 + 1e-8f);
    const float r1 = 1.0f / (d1 + 1e-8f);
    const float r2 = 1.0f / (d2 + 1e-8f);
    const float rs = 1.0f / (r0 + r1 + r2);
    const size_t o = ((size_t)b * S_ + s) * 3;
    wbuf[o + 0] = r0 * rs;
    wbuf[o + 1] = r1 * rs;
    wbuf[o + 2] = r2 * rs;
    ibuf[o + 0] = i0;
    ibuf[o + 1] = i1;
    ibuf[o + 2] = i2;
  }
}

// ---------------------------------------------------------------------------
// Kernel 2: tiled transpose points1 [B,D,N] -> ptT [B,N,D] so the 3-NN
// gather becomes contiguous 1 KB row reads.
// ---------------------------------------------------------------------------
__global__ __launch_bounds__(256)
void transpose_kernel(const float* __restrict__ in, float* __restrict__ out) {
  __shared__ float tile[32][33];
  const int b   = blockIdx.z;
  const int n0  = blockIdx.x * 32;
  const int dd0 = blockIdx.y * 32;
  const int tx  = threadIdx.x;   // 0..31
  const int ty  = threadIdx.y;   // 0..7
  const float* inb  = in  + (size_t)b * D_ * N_;
  float*       outb = out + (size_t)b * N_ * D_;

#pragma unroll
  for (int i = 0; i < 32; i += 8)
    tile[ty + i][tx] = inb[(size_t)(dd0 + ty + i) * N_ + (n0 + tx)];
  __syncthreads();
#pragma unroll
  for (int i = 0; i < 32; i += 8)
    outb[(size_t)(n0 + ty + i) * D_ + (dd0 + tx)] = tile[tx][ty + i];
}

// ---------------------------------------------------------------------------
// Kernel 3: weighted 3-NN interpolation.  Block = (batch b, 16 queries).
// Phase 1: each thread owns one feature d, reads 3 neighbor rows
// (contiguous when transposed: idx_mul=D_, d_mul=1), fma's, stores the
// query's row into LDS.  Phase 2: write out[b,d,s] with s-contiguous bursts.
// acc pitch 257 (odd) keeps the transposed read phase bank-conflict-free.
// ---------------------------------------------------------------------------
__global__ __launch_bounds__(256)
void interp_kernel(const float* __restrict__ src,
                   const float* __restrict__ wbuf,
                   const int* __restrict__ ibuf,
                   float* __restrict__ out,
                   int idx_mul, int d_mul) {
  __shared__ float acc[16 * 257];
  const int s0  = blockIdx.x * 16;
  const int b   = blockIdx.y;
  const int tid = threadIdx.x;

  const float* srcb = src + (size_t)b * N_ * D_;   // same extent either layout

  for (int sl = 0; sl < 16; ++sl) {
    const size_t q = ((size_t)b * S_ + s0 + sl) * 3;
    const float w0 = wbuf[q + 0], w1 = wbuf[q + 1], w2 = wbuf[q + 2];
    const int   j0 = ibuf[q + 0], j1 = ibuf[q + 1], j2 = ibuf[q + 2];
    const int   d  = tid;
    const float r = w0 * srcb[(size_t)j0 * idx_mul + (size_t)d * d_mul]
                  + w1 * srcb[(size_t)j1 * idx_mul + (size_t)d * d_mul]
                  + w2 * srcb[(size_t)j2 * idx_mul + (size_t)d * d_mul];
    acc[sl * 257 + d] = r;
  }
  __syncthreads();

  const int sx = tid & 15;        // s within tile -> contiguous output
  const int ty = tid >> 4;        // d sub-index
#pragma unroll
  for (int pass = 0; pass < 16; ++pass) {
    const int d = pass * 16 + ty;
    out[((size_t)b * D_ + d) * S_ + (s0 + sx)] = acc[sx * 257 + d];
  }
}

// ---------------------------------------------------------------------------
extern "C" void kernel_launch(void* const* d_in, const int* in_sizes, int n_in,
                              void* d_out, int out_size, void* d_ws, size_t ws_size,
                              hipStream_t stream) {
  const float* xyz1    = (const float*)d_in[0];  // [B,3,N]
  const float* xyz2    = (const float*)d_in[1];  // [B,3,S]
  const float* points1 = (const float*)d_in[2];  // [B,D,N]
  float* out = (float*)d_out;                    // [B,D,S]

  const size_t wbytes = (size_t)B_ * S_ * 3 * sizeof(float);
  float* wbuf = (float*)d_ws;
  int*   ibuf = (int*)((char*)d_ws + wbytes);
  float* ptT  = (float*)((char*)d_ws + 2 * wbytes);
  const size_t need_transposed = 2 * wbytes + (size_t)B_ * N_ * D_ * sizeof(float);
  const bool useT = (ws_size >= need_transposed);

  // 1) WMMA distance + top-3 + weights.  128 blocks x 8 waves x 16 queries.
  knn3_wmma_kernel<<<dim3(B_ * S_ / 128), dim3(256), 0, stream>>>(
      xyz1, xyz2, wbuf, ibuf);

  // 2) Optional layout transpose for coalesced gathers.
  const float* src = points1;
  int idx_mul = 1, d_mul = N_;
  if (useT) {
    transpose_kernel<<<dim3(N_ / 32, D_ / 32, B_), dim3(32, 8), 0, stream>>>(
        points1, ptT);
    src = ptT; idx_mul = D_; d_mul = 1;
  }

  // 3) Weighted gather-interpolate + transposed writeout.
  interp_kernel<<<dim3(S_ / 16, B_), dim3(256), 0, stream>>>(
      src, wbuf, ibuf, out, idx_mul, d_mul);
}